// PatchNetVLAD_37898791420459
// MI455X (gfx1250) — compile-verified
//
#include <hip/hip_runtime.h>

#define BB 4
#define CC 512
#define HWSZ 784           // 28*28, == 49*16, divisible by 4
#define KK 64
#define HW4 (HWSZ / 4)     // 196 float4 per row

typedef float v2f __attribute__((ext_vector_type(2)));
typedef float v4f __attribute__((ext_vector_type(4)));
typedef float v8f __attribute__((ext_vector_type(8)));

// ---------------------------------------------------------------------------
// Kernel 1: soft_assign[b,k,hw] = softmax_k( conv_w[k,:] . x[b,:,hw] )
// Grid: B * 49 blocks (one 16-pixel tile each), 128 threads = 4 waves.
// Wave w computes the 16x16 tile for k-rows [16w, 16w+16) with fp32 WMMA,
// accumulating over C=512 in steps of 4 (V_WMMA_F32_16X16X4_F32).
// ---------------------------------------------------------------------------
__global__ __launch_bounds__(128) void vlad_assign_kernel(
    const float* __restrict__ x,       // [B, C, HW]
    const float* __restrict__ conv_w,  // [K, C]
    float* __restrict__ sa)            // [B, K, HW]  (workspace)
{
    __shared__ float sa_lds[KK][16];

    const int tile = blockIdx.x;
    const int b    = tile / (HWSZ / 16);
    const int hw0  = (tile % (HWSZ / 16)) * 16;
    const int tid  = threadIdx.x;
    const int wave = tid >> 5;
    const int lane = tid & 31;
    const int half = lane >> 4;   // 0: lanes 0-15, 1: lanes 16-31
    const int l16  = lane & 15;

    const float* xb   = x + (size_t)b * CC * HWSZ;
    const float* arow = conv_w + (size_t)(wave * 16 + l16) * CC;  // A row, M = l16

    v8f acc = {};
    for (int c0 = 0; c0 < CC; c0 += 4) {
        const int kk = c0 + half * 2;   // lane-half selects K pair {0,1} vs {2,3}
        v2f a;
        a.x = arow[kk];
        a.y = arow[kk + 1];
        v2f bv;
        bv.x = xb[(size_t)kk * HWSZ + hw0 + l16];
        bv.y = xb[(size_t)(kk + 1) * HWSZ + hw0 + l16];
        // D = A(16x4, f32) * B(4x16, f32) + C   -> v_wmma_f32_16x16x4_f32
        acc = __builtin_amdgcn_wmma_f32_16x16x4_f32(
            /*neg_a=*/false, a, /*neg_b=*/false, bv,
            /*c_mod=*/(short)0, acc, /*reuse_a=*/false, /*reuse_b=*/false);
    }

    // D layout: acc[r] on this lane is (M = r + 8*half, N = l16) of the tile.
#pragma unroll
    for (int r = 0; r < 8; ++r)
        sa_lds[wave * 16 + half * 8 + r][l16] = acc[r];
    __syncthreads();

    // Softmax over K=64 per pixel; 16 lanes each own one pixel column.
    if (tid < 16) {
        const int p = tid;
        float m = sa_lds[0][p];
#pragma unroll 8
        for (int k = 1; k < KK; ++k) m = fmaxf(m, sa_lds[k][p]);
        float s = 0.0f;
#pragma unroll 8
        for (int k = 0; k < KK; ++k) {
            const float e = __expf(sa_lds[k][p] - m);
            sa_lds[k][p] = e;
            s += e;
        }
        const float inv = 1.0f / s;
#pragma unroll 8
        for (int k = 0; k < KK; ++k)
            sa[((size_t)b * KK + k) * HWSZ + hw0 + p] = sa_lds[k][p] * inv;
    }
}

// ---------------------------------------------------------------------------
// Kernel 2: out[b,k,c,hw] = (x[b,c,hw] - centroids[0][c]) * sa[b,k,hw]
// Pure streaming writer: 411 MB of stores is the roofline (~17.6 us @ 23.3TB/s).
// float4 per thread, grid-stride, coalesced; non-temporal stores so the huge
// output stream does not evict the L2-resident x / sa inputs.
// ---------------------------------------------------------------------------
__global__ __launch_bounds__(256) void vlad_store_kernel(
    const float* __restrict__ x,          // [B, C, HW]
    const float* __restrict__ centroids,  // [K, C] (only row 0 used)
    const float* __restrict__ sa,         // [B, K, HW]
    float* __restrict__ out,              // [B, K, C, HW]
    int total4)
{
    const int stride = gridDim.x * blockDim.x;
    for (int i = blockIdx.x * blockDim.x + threadIdx.x; i < total4; i += stride) {
        const int hw4 = i % HW4;       // float4 index within a (b,k,c) row
        const int r   = i / HW4;
        const int c   = r % CC;
        const int r2  = r / CC;
        const int k   = r2 % KK;
        const int b   = r2 / KK;

        const v4f xv = *(const v4f*)(x  + ((size_t)(b * CC + c)) * HWSZ + hw4 * 4);
        const v4f sv = *(const v4f*)(sa + ((size_t)(b * KK + k)) * HWSZ + hw4 * 4);
        const float ct = centroids[c];   // centroids[0][c]

        v4f o;
        o.x = (xv.x - ct) * sv.x;
        o.y = (xv.y - ct) * sv.y;
        o.z = (xv.z - ct) * sv.z;
        o.w = (xv.w - ct) * sv.w;
        __builtin_nontemporal_store(o, (v4f*)(out + (size_t)i * 4));
    }
}

extern "C" void kernel_launch(void* const* d_in, const int* in_sizes, int n_in,
                              void* d_out, int out_size, void* d_ws, size_t ws_size,
                              hipStream_t stream) {
    const float* x         = (const float*)d_in[0];  // [4,512,28,28]
    const float* conv_w    = (const float*)d_in[1];  // [64,512]
    const float* centroids = (const float*)d_in[2];  // [64,512]
    float* out = (float*)d_out;                      // [4,64,512,28,28]
    float* sa  = (float*)d_ws;                       // needs 4*64*784*4 = 784 KB

    // Phase 1: WMMA GEMM + softmax -> soft_assign in workspace.
    vlad_assign_kernel<<<BB * (HWSZ / 16), 128, 0, stream>>>(x, conv_w, sa);

    // Phase 2: bandwidth-bound broadcast-multiply writer.
    const int total4 = (BB * KK * CC * HWSZ) / 4;    // 25,690,112 float4 stores
    vlad_store_kernel<<<4096, 256, 0, stream>>>(x, centroids, sa, out, total4);
}